// GcnNet_88897233092952
// MI455X (gfx1250) — compile-verified
//
#include <hip/hip_runtime.h>

typedef float v2f __attribute__((ext_vector_type(2)));
typedef float v8f __attribute__((ext_vector_type(8)));

#define CH 300      // channels
#define CHPAD 304   // padded #rows of W copy (multiple of 16)
#define TP 4        // pool dim

// ---------------- utility kernels ----------------

__global__ void fill_kernel(float* __restrict__ p, float v, int n) {
    int i = blockIdx.x * blockDim.x + threadIdx.x;
    if (i < n) p[i] = v;
}

__global__ void deg_kernel(const int* __restrict__ row, float* __restrict__ deg, int E) {
    int e = blockIdx.x * blockDim.x + threadIdx.x;
    if (e < E) unsafeAtomicAdd(&deg[row[e]], 1.0f);   // global_atomic_add_f32
}

__global__ void rsqrt_kernel(float* __restrict__ d, int n) {
    int i = blockIdx.x * blockDim.x + threadIdx.x;
    if (i < n) d[i] = rsqrtf(d[i]);
}

// Wp[0:90000] = W, Wp[90000:91200] = 0  (rows 300..303 zero-padded)
__global__ void padw_kernel(const float* __restrict__ W, float* __restrict__ Wp) {
    int i = blockIdx.x * blockDim.x + threadIdx.x;
    if (i < CHPAD * CH) Wp[i] = (i < CH * CH) ? W[i] : 0.0f;
}

// xm[n,c] = mean_t x[n,t,c]
__global__ void mean_kernel(const float* __restrict__ x, float* __restrict__ xm, int total) {
    int i = blockIdx.x * blockDim.x + threadIdx.x;
    if (i >= total) return;
    int n = i / CH;
    int c = i - n * CH;
    const float* b = x + (size_t)n * (TP * CH) + c;
    xm[i] = 0.25f * (b[0] + b[CH] + b[2 * CH] + b[3 * CH]);
}

// ---------------- WMMA GEMM: C[M,300] = A[M,300] * Wp^T + bias ----------------
// 4 waves/block; each wave computes a 16(row) x 64(col) tile with f32 WMMA 16x16x4.
// A frag (32-bit A 16x4): lane = m + 16*(k>=2), vgpr = k%2 -> float2 A[row0+m][k0+2*half]
// B frag (4x16 K x N):    lane = n + 16*(k>=2), vgpr = k%2 ; B[k][n] = Wp[n][k]
// C/D (16x16 f32): vgpr j, lanes 0-15: M=j, N=lane; lanes 16-31: M=j+8, N=lane-16
// Wp is zero-padded to CHPAD rows so the K-loop is completely branchless; the
// 4 N-subtiles share one base pointer via constant 19200B instruction offsets.
__global__ __launch_bounds__(128) void gemm_wmma_kernel(const float* __restrict__ A,
                                                        const float* __restrict__ Wp,
                                                        const float* __restrict__ bias,
                                                        float* __restrict__ C,
                                                        int mtiles) {
    const int wave = threadIdx.x >> 5;
    const int lane = threadIdx.x & 31;
    const int trow = blockIdx.x * 4 + wave;
    if (trow >= mtiles) return;                 // uniform per wave (scalar branch)

    const int mn   = lane & 15;
    const int half = lane >> 4;
    const int row0 = trow * 16;
    const int col0 = blockIdx.y * 64;

    v8f acc[4] = {};

    const float* ap = A  + (size_t)(row0 + mn) * CH + 2 * half;
    const float* bp = Wp + (size_t)(col0 + mn) * CH + 2 * half;

    for (int k0 = 0; k0 < CH; k0 += 4) {
        v2f a = *(const v2f*)(ap + k0);
#pragma unroll
        for (int t = 0; t < 4; ++t) {
            v2f b = *(const v2f*)(bp + k0 + t * (16 * CH));   // const 19200B offsets
            acc[t] = __builtin_amdgcn_wmma_f32_16x16x4_f32(
                false, a, false, b, (short)0, acc[t], false, false);
        }
    }

#pragma unroll
    for (int t = 0; t < 4; ++t) {
        int n = col0 + t * 16 + mn;
        if (n < CH) {
            float bv = bias[n];
#pragma unroll
            for (int j = 0; j < 8; ++j) {
                int m = row0 + j + 8 * half;
                C[(size_t)m * CH + n] = acc[t][j] + bv;
            }
        }
    }
}

// ---------------- edge scatter: acc[col[e], c] += dis[row[e]] * h[row[e], c] ----------------
__global__ void scatter_kernel(const int* __restrict__ row, const int* __restrict__ col,
                               const float* __restrict__ dis, const float* __restrict__ h,
                               float* __restrict__ acc) {
    int e = blockIdx.x;
    int c = blockIdx.y * blockDim.x + threadIdx.x;
    if (c >= CH) return;
    int r = row[e];
    int d = col[e];
    float v = dis[r] * h[(size_t)r * CH + c];
    unsafeAtomicAdd(&acc[(size_t)d * CH + c], v);
}

// out[i,c] = act( dis[i] * (acc[i,c] + dis[i]*h[i,c]) )   (self-loop folded in)
__global__ void finalize_kernel(const float* __restrict__ acc, const float* __restrict__ h,
                                const float* __restrict__ dis, float* __restrict__ out,
                                int total, int relu) {
    int i = blockIdx.x * blockDim.x + threadIdx.x;
    if (i >= total) return;
    int n = i / CH;
    float dn = dis[n];
    float v = dn * (acc[i] + dn * h[i]);
    if (relu) v = (v > 0.0f) ? v : 0.01f * v;
    out[i] = v;
}

// ---------------- launcher ----------------
extern "C" void kernel_launch(void* const* d_in, const int* in_sizes, int n_in,
                              void* d_out, int out_size, void* d_ws, size_t ws_size,
                              hipStream_t stream) {
    const float* x  = (const float*)d_in[0];
    const int*   ei = (const int*)d_in[1];
    const float* W1 = (const float*)d_in[2];
    const float* b1 = (const float*)d_in[3];
    const float* W2 = (const float*)d_in[4];
    const float* b2 = (const float*)d_in[5];
    float* out = (float*)d_out;

    const int E = in_sizes[1] / 2;
    const int N = in_sizes[0] / (TP * CH);
    const int total = N * CH;
    const int* row = ei;
    const int* col = ei + E;

    char* ws = (char*)d_ws;
    float* dis = (float*)ws;                       // N floats
    size_t off = ((size_t)N * sizeof(float) + 255) & ~(size_t)255;
    float* Wp = (float*)(ws + off);                // CHPAD*CH floats (zero-padded W copy)
    off += (size_t)CHPAD * CH * sizeof(float);
    float* bufA = (float*)(ws + off);              // xm, then h1
    float* bufB = bufA + total;                    // h0, then h2
    float* bufC = bufB + total;                    // acc of layer 1

    const int TB = 256;
    const int mtiles = (N + 15) / 16;
    dim3 gg((mtiles + 3) / 4, (CH + 63) / 64);
    dim3 gs(E, (CH + 127) / 128);
    const int padBlocks = (CHPAD * CH + TB - 1) / TB;

    // degree (self loop contributes 1.0 at init) -> dis = deg^-1/2
    fill_kernel<<<(N + TB - 1) / TB, TB, 0, stream>>>(dis, 1.0f, N);
    deg_kernel<<<(E + TB - 1) / TB, TB, 0, stream>>>(row, dis, E);
    rsqrt_kernel<<<(N + TB - 1) / TB, TB, 0, stream>>>(dis, N);

    // mean pool (commutes with linear layer)
    mean_kernel<<<(total + TB - 1) / TB, TB, 0, stream>>>(x, bufA, total);

    // layer 1 GEMM + bias
    padw_kernel<<<padBlocks, TB, 0, stream>>>(W1, Wp);
    gemm_wmma_kernel<<<gg, 128, 0, stream>>>(bufA, Wp, b1, bufB, mtiles);

    // propagate layer 1
    fill_kernel<<<(total + TB - 1) / TB, TB, 0, stream>>>(bufC, 0.0f, total);
    scatter_kernel<<<gs, 128, 0, stream>>>(row, col, dis, bufB, bufC);
    finalize_kernel<<<(total + TB - 1) / TB, TB, 0, stream>>>(bufC, bufB, dis, bufA, total, 1);

    // layer 2 GEMM + bias
    padw_kernel<<<padBlocks, TB, 0, stream>>>(W2, Wp);
    gemm_wmma_kernel<<<gg, 128, 0, stream>>>(bufA, Wp, b2, bufB, mtiles);

    // propagate layer 2 directly into d_out
    fill_kernel<<<(total + TB - 1) / TB, TB, 0, stream>>>(out, 0.0f, total);
    scatter_kernel<<<gs, 128, 0, stream>>>(row, col, dis, bufB, out);
    finalize_kernel<<<(total + TB - 1) / TB, TB, 0, stream>>>(out, bufB, dis, out, total, 0);
}